// YOLOLoss_86414741996255
// MI455X (gfx1250) — compile-verified
//
#include <hip/hip_runtime.h>
#include <hip/hip_bf16.h>
#include <math.h>

// Problem constants (fixed by setup_inputs): pred (B,A,H,W,5+C)
#define BB 32
#define AA 3
#define HH 64
#define WW 64
#define CC 80
#define NN 2048
#define CH (5 + CC)
#define CELLS (BB * AA * HH * WW)

typedef __attribute__((ext_vector_type(2))) float v2f;
typedef __attribute__((ext_vector_type(8))) float v8f;

// bce(x, t) = max(x,0) - x*t + log1p(exp(-|x|));  g(x) is the t-independent part.
__device__ __forceinline__ float bce_g(float x) {
    return fmaxf(x, 0.0f) + log1pf(expf(-fabsf(x)));
}

// ---------------------------------------------------------------------------
// Kernel 1: init workspace (accumulators = 0, obj map = 0, kill map = 1)
// acc[0]=xywh_sum acc[1]=cls_sum acc[2]=conf_sum acc[3]=n_obj acc[4]=n_noobj
// ---------------------------------------------------------------------------
__global__ void k_init(float* __restrict__ objm, float* __restrict__ killm,
                       float* __restrict__ acc) {
    int i = blockIdx.x * blockDim.x + threadIdx.x;
    if (i < 8) acc[i] = 0.0f;
    for (; i < CELLS; i += gridDim.x * blockDim.x) {
        objm[i]  = 0.0f;
        killm[i] = 1.0f;
    }
}

// ---------------------------------------------------------------------------
// Kernel 2: per-target work. IoU argmax over 3 anchors (first-max-on-tie),
// scatter obj / noobj-kill maps, bbox MSE partial, -x[label] cls partial,
// and record the flat pred cell base for the WMMA class-sum kernel.
// ---------------------------------------------------------------------------
__global__ void k_target(const float* __restrict__ pred,
                         const float* __restrict__ target,
                         const float* __restrict__ anchors,
                         float* __restrict__ acc,
                         int*   __restrict__ cellbase,
                         float* __restrict__ objm,
                         float* __restrict__ killm) {
    const int n = blockIdx.x * blockDim.x + threadIdx.x;
    if (n >= NN) return;

    const float tb  = target[n * 6 + 0];
    const float tl  = target[n * 6 + 1];
    const float xy0 = target[n * 6 + 2] * (float)HH;
    const float xy1 = target[n * 6 + 3] * (float)HH;
    const float wh0 = target[n * 6 + 4] * (float)HH;
    const float wh1 = target[n * 6 + 5] * (float)HH;

    const int b     = (int)tb;
    const int label = (int)tl;
    const int wi    = (int)wh0;   // NOTE: reference indexes the grid with wh
    const int hi    = (int)wh1;

    // IoU vs the 3 anchors; argmax keeps the first maximum (jnp.argmax ties)
    float iou_a[AA];
    float best = -1.0f;
    int   bi   = 0;
#pragma unroll
    for (int a = 0; a < AA; ++a) {
        const float aw = anchors[2 * a + 0];
        const float ah = anchors[2 * a + 1];
        const float inter = fminf(aw, wh0) * fminf(ah, wh1);
        const float iou   = inter / (aw * ah + wh0 * wh1 - inter);
        iou_a[a] = iou;
        if (iou > best) { best = iou; bi = a; }
    }

    const int cell = ((b * AA + bi) * HH + hi) * WW + wi;
    objm[cell] = 1.0f;  // duplicate scatter-set of the same value: benign race
#pragma unroll
    for (int a = 0; a < AA; ++a) {
        // keep = (iou <= 0.5); scatter-mul by keep == write 0 when iou > 0.5
        if (iou_a[a] > 0.5f)
            killm[((b * AA + a) * HH + hi) * WW + wi] = 0.0f;
    }

    const int base = cell * CH;
    cellbase[n] = base;

    // loss_xywh contribution
    const float tx = xy0 - floorf(xy0);
    const float ty = xy1 - floorf(xy1);
    const float tw = logf(wh0 / anchors[2 * bi + 0]);
    const float th = logf(wh1 / anchors[2 * bi + 1]);
    const float d0 = pred[base + 0] - tx;
    const float d1 = pred[base + 1] - ty;
    const float d2 = pred[base + 2] - tw;
    const float d3 = pred[base + 3] - th;
    atomicAdd(&acc[0], d0 * d0 + d1 * d1 + d2 * d2 + d3 * d3);

    // one-hot term of loss_cls:  -x[n, label]
    atomicAdd(&acc[1], -pred[base + 5 + label]);
}

// ---------------------------------------------------------------------------
// Kernel 3: class-loss row sums via V_WMMA_F32_16X16X4_F32.
// One wave32 per 16-target tile.  D = A(16x4) x Ones(4x16) + C  accumulated
// over 20 K-chunks (K=80).  With all-ones B, D[m][n] = rowsum(m) for all n,
// so lane 0 (M=0..7) + lane 16 (M=8..15) hold the tile total between them.
// A layout: lanes 0-15 -> M=lane, K={0,1}; lanes 16-31 -> M=lane-16, K={2,3}.
// EXEC is all-1s: N=2048 is exactly 128 full waves, no divergence here.
// ---------------------------------------------------------------------------
__global__ void k_cls(const float* __restrict__ pred,
                      const int*   __restrict__ cellbase,
                      float* __restrict__ acc) {
    const int tid  = blockIdx.x * blockDim.x + threadIdx.x;
    const int wave = tid >> 5;
    const int lane = tid & 31;
    const int m    = lane & 15;
    const int koff = (lane < 16) ? 0 : 2;
    const int base = cellbase[wave * 16 + m] + 5;

    v8f c = {};
    v2f bones;
    bones.x = 1.0f;
    bones.y = 1.0f;

#pragma unroll
    for (int kk = 0; kk < CC; kk += 4) {
        const float x0 = pred[base + kk + koff + 0];
        const float x1 = pred[base + kk + koff + 1];
        v2f a;
        a.x = bce_g(x0);
        a.y = bce_g(x1);
        c = __builtin_amdgcn_wmma_f32_16x16x4_f32(
                /*neg_a=*/false, a, /*neg_b=*/false, bones,
                /*c_mod=*/(short)0, c, /*reuse_a=*/false, /*reuse_b=*/false);
    }

    const float s = c[0] + c[1] + c[2] + c[3] + c[4] + c[5] + c[6] + c[7];
    if (lane == 0 || lane == 16) atomicAdd(&acc[1], s);
}

// ---------------------------------------------------------------------------
// Kernel 4: confidence loss over the full (B,A,H,W) grid.  Strided channel-4
// read (the bandwidth-dominant stream), wave32 shuffle reduction, 3 atomics
// per wave.
// ---------------------------------------------------------------------------
__global__ void k_conf(const float* __restrict__ pred,
                       const float* __restrict__ objm,
                       const float* __restrict__ killm,
                       float* __restrict__ acc) {
    const int stride = gridDim.x * blockDim.x;
    float lo = 0.0f, ln = 0.0f, lc = 0.0f;
    for (int i = blockIdx.x * blockDim.x + threadIdx.x; i < CELLS; i += stride) {
        if (i + stride < CELLS)
            __builtin_prefetch(pred + (size_t)(i + stride) * CH + 4, 0, 1);
        const float obj   = objm[i];
        const float noobj = (1.0f - obj) * killm[i];
        const float x = pred[(size_t)i * CH + 4];
        const float g = bce_g(x);
        lc += obj * (g - x) + noobj * g;   // t=1: g-x ; t=0: g
        lo += obj;
        ln += noobj;
    }
#pragma unroll
    for (int off = 16; off > 0; off >>= 1) {
        lo += __shfl_down(lo, off);
        ln += __shfl_down(ln, off);
        lc += __shfl_down(lc, off);
    }
    if ((threadIdx.x & 31) == 0) {
        atomicAdd(&acc[2], lc);
        atomicAdd(&acc[3], lo);
        atomicAdd(&acc[4], ln);
    }
}

// ---------------------------------------------------------------------------
// Kernel 5: combine into the scalar loss.
// ---------------------------------------------------------------------------
__global__ void k_final(const float* __restrict__ acc, float* __restrict__ out) {
    if (blockIdx.x == 0 && threadIdx.x == 0) {
        const float loss_xywh = acc[0] / (float)(NN * 4);
        const float loss_cls  = acc[1] / (float)(NN * CC);
        const float loss_conf = acc[2] / (acc[3] + acc[4]);
        out[0] = loss_xywh + loss_conf + loss_cls;
    }
}

extern "C" void kernel_launch(void* const* d_in, const int* in_sizes, int n_in,
                              void* d_out, int out_size, void* d_ws, size_t ws_size,
                              hipStream_t stream) {
    const float* pred    = (const float*)d_in[0];
    const float* target  = (const float*)d_in[1];
    const float* anchors = (const float*)d_in[2];
    // d_in[3] = num_classes scalar (hardcoded CC=80)
    float* out = (float*)d_out;

    // Workspace layout (all 4-byte aligned from base):
    float* acc      = (float*)d_ws;            // 8 floats of accumulators
    int*   cellbase = (int*)(acc + 8);         // NN ints
    float* objm     = (float*)(cellbase + NN); // CELLS floats
    float* killm    = objm + CELLS;            // CELLS floats

    k_init  <<<1536, 256, 0, stream>>>(objm, killm, acc);
    k_target<<<NN / 256, 256, 0, stream>>>(pred, target, anchors, acc, cellbase,
                                           objm, killm);
    k_cls   <<<(NN / 16) * 32 / 256, 256, 0, stream>>>(pred, cellbase, acc);
    k_conf  <<<256, 256, 0, stream>>>(pred, objm, killm, acc);
    k_final <<<1, 1, 0, stream>>>(acc, out);
}